// LeafSelectionHead_11776800326351
// MI455X (gfx1250) — compile-verified
//
#include <hip/hip_runtime.h>
#include <hip/hip_bf16.h>

#define Gg 512
#define NPER 200
#define MAXN 256
#define IN_DIM 128
#define HID 64
#define Nn (Gg * NPER)          // 102400
#define Ee (Nn * 8)             // 819200
#define LDS_STRIDE 132          // padded row stride (floats): conflict-free b64 reads

typedef float v2f __attribute__((ext_vector_type(2)));
typedef float v8f __attribute__((ext_vector_type(8)));

__device__ __forceinline__ int gtid() { return blockIdx.x * blockDim.x + threadIdx.x; }

// ---- init: deg=1.0 (self loop), xpool=0, y_leaf = NaN --------------------
__global__ void k_init(float* __restrict__ deg, float* __restrict__ xpool,
                       float* __restrict__ yleaf) {
  int t = gtid();
  if (t < Nn)            deg[t]   = 1.0f;
  if (t < Gg * HID)      xpool[t] = 0.0f;
  if (t < Gg * MAXN * 2) yleaf[t] = __int_as_float(0x7FC00000); // quiet NaN
}

// ---- in-degree count at dst ----------------------------------------------
__global__ void k_deg(const int* __restrict__ dst, float* __restrict__ deg) {
  int e = gtid(); if (e >= Ee) return;
  atomicAdd(deg + dst[e], 1.0f);
}

// ---- dinv = rsqrt(deg), in place -----------------------------------------
__global__ void k_dinv(float* __restrict__ deg) {
  int n = gtid(); if (n >= Nn) return;
  deg[n] = rsqrtf(deg[n]);
}

// ---- fused GEMM + self-loop epilogue via V_WMMA_F32_16X16X4_F32 ----------
// Block stages W^T in LDS once (33 KB, padded stride 132 -> bank-conflict-free
// ds_load_b64 for B fragments). One wave owns a 16-row strip of X and computes
// all four 16x16 col tiles: A fragment loaded once per k-step from global,
// shared across 4 WMMAs (128 WMMAs/wave); B fragments served from LDS.
// A frag (16x4): lane l holds X [row0 + l%16, k + 2*(l/16) + v], v=0..1
// B frag (4x16): lane l holds W [k + 2*(l/16) + v, 16*ct + l%16]
// C/D:           VGPR v, lane l -> C[row0 + 8*(l/16) + v, 16*ct + l%16]
// Epilogue writes h_xw and agg = dinv(row)^2 * h_xw + b_body[col].
__global__ void k_gemm_wmma(const float* __restrict__ X, const float* __restrict__ W,
                            const float* __restrict__ dinv, const float* __restrict__ bias,
                            float* __restrict__ hxw, float* __restrict__ agg) {
  __shared__ float lw[HID * LDS_STRIDE];     // lw[n][kk] = W[kk][n]
  for (int t = threadIdx.x; t < IN_DIM * HID; t += blockDim.x) {
    int kk = t >> 6, n = t & 63;             // coalesced read of W[kk][n]
    lw[n * LDS_STRIDE + kk] = W[t];
  }
  __syncthreads();

  int wave = blockIdx.x * (blockDim.x >> 5) + (threadIdx.x >> 5);
  int lane = threadIdx.x & 31;
  int row0 = wave * 16;
  if (row0 >= Nn) return;                    // uniform per wave (grid is exact)
  int g  = lane >> 4;                        // lane group 0/1
  int lm = lane & 15;

  v8f acc0 = {}, acc1 = {}, acc2 = {}, acc3 = {};
  const float* __restrict__ xrow = X + (size_t)(row0 + lm) * IN_DIM + 2 * g;
  const float* lw0 = lw + ( 0 + lm) * LDS_STRIDE + 2 * g;
  const float* lw1 = lw + (16 + lm) * LDS_STRIDE + 2 * g;
  const float* lw2 = lw + (32 + lm) * LDS_STRIDE + 2 * g;
  const float* lw3 = lw + (48 + lm) * LDS_STRIDE + 2 * g;
#pragma unroll
  for (int k = 0; k < IN_DIM; k += 4) {
    v2f a  = *(const v2f*)(xrow + k);
    v2f b0 = *(const v2f*)(lw0 + k);
    v2f b1 = *(const v2f*)(lw1 + k);
    v2f b2 = *(const v2f*)(lw2 + k);
    v2f b3 = *(const v2f*)(lw3 + k);
    acc0 = __builtin_amdgcn_wmma_f32_16x16x4_f32(false, a, false, b0, (short)0, acc0, false, false);
    acc1 = __builtin_amdgcn_wmma_f32_16x16x4_f32(false, a, false, b1, (short)0, acc1, false, false);
    acc2 = __builtin_amdgcn_wmma_f32_16x16x4_f32(false, a, false, b2, (short)0, acc2, false, false);
    acc3 = __builtin_amdgcn_wmma_f32_16x16x4_f32(false, a, false, b3, (short)0, acc3, false, false);
  }
  float bc0 = bias[lm +  0], bc1 = bias[lm + 16];
  float bc2 = bias[lm + 32], bc3 = bias[lm + 48];
#pragma unroll
  for (int v = 0; v < 8; ++v) {
    int row = row0 + g * 8 + v;
    float di = dinv[row];
    float d2 = di * di;
    size_t base = (size_t)row * HID + lm;
    float c0 = acc0[v], c1 = acc1[v], c2 = acc2[v], c3 = acc3[v];
    hxw[base +  0] = c0;  agg[base +  0] = d2 * c0 + bc0;
    hxw[base + 16] = c1;  agg[base + 16] = d2 * c1 + bc1;
    hxw[base + 32] = c2;  agg[base + 32] = d2 * c2 + bc2;
    hxw[base + 48] = c3;  agg[base + 48] = d2 * c3 + bc3;
  }
}

// ---- edge scatter-add: agg[dst] += dinv[src]*dinv[dst]*h_xw[src] ---------
// 16 threads per edge, float4 per thread (coalesced row gather).
__global__ void k_edge64(const float* __restrict__ hxw, const float* __restrict__ dinv,
                         const int* __restrict__ src, const int* __restrict__ dst,
                         float* __restrict__ agg) {
  int t = gtid();
  int e = t >> 4; if (e >= Ee) return;
  int part = t & 15;
  int s = src[e], d = dst[e];
  float w = dinv[s] * dinv[d];
  float4 hv = ((const float4*)(hxw + (size_t)s * HID))[part];
  float* ap = agg + (size_t)d * HID + part * 4;
  atomicAdd(ap + 0, w * hv.x);
  atomicAdd(ap + 1, w * hv.y);
  atomicAdd(ap + 2, w * hv.z);
  atomicAdd(ap + 3, w * hv.w);
}

// ---- h = leaky_relu(agg) in place; pool h into xpool[batch] --------------
__global__ void k_leaky_pool(float* __restrict__ agg, const int* __restrict__ batch,
                             float* __restrict__ xpool) {
  int t = gtid(); if (t >= Nn * HID) return;
  int n = t >> 6, j = t & 63;
  float v = agg[t];
  v = (v >= 0.0f) ? v : 0.01f * v;
  agg[t] = v;
  atomicAdd(xpool + (size_t)batch[n] * HID + j, v);
}

// ---- leaf: tvec = h @ W_leaf [N,2]; agg2 = dinv^2*tvec + b_leaf ----------
__global__ void k_leaf_xw(const float* __restrict__ h, const float* __restrict__ Wl,
                          const float* __restrict__ bl, const float* __restrict__ dinv,
                          float* __restrict__ tvec, float* __restrict__ agg2) {
  int i = gtid(); if (i >= Nn * 2) return;
  int n = i >> 1, c = i & 1;
  const float* hr = h + (size_t)n * HID;
  float s = 0.0f;
#pragma unroll
  for (int j = 0; j < HID; ++j) s += hr[j] * Wl[j * 2 + c];
  tvec[i] = s;
  float di = dinv[n];
  agg2[i] = di * di * s + bl[c];
}

// ---- leaf edge scatter-add -----------------------------------------------
__global__ void k_edge2(const float* __restrict__ tvec, const float* __restrict__ dinv,
                        const int* __restrict__ src, const int* __restrict__ dst,
                        float* __restrict__ agg2) {
  int e = gtid(); if (e >= Ee) return;
  int s = src[e], d = dst[e];
  float w = dinv[s] * dinv[d];
  atomicAdd(agg2 + (size_t)d * 2 + 0, w * tvec[(size_t)s * 2 + 0]);
  atomicAdd(agg2 + (size_t)d * 2 + 1, w * tvec[(size_t)s * 2 + 1]);
}

// ---- scatter leaf logits into NaN-padded [G, MAXN, 2] --------------------
__global__ void k_scatter(const float* __restrict__ agg2, const int* __restrict__ batch,
                          const int* __restrict__ kpos, float* __restrict__ yleaf) {
  int n = gtid(); if (n >= Nn) return;
  int g = batch[n], kk = kpos[n];
  size_t o = ((size_t)g * MAXN + kk) * 2;
  yleaf[o + 0] = agg2[(size_t)n * 2 + 0];
  yleaf[o + 1] = agg2[(size_t)n * 2 + 1];
}

// ---- y_eos = xpool @ W_eos + b_eos ---------------------------------------
__global__ void k_eos(const float* __restrict__ xpool, const float* __restrict__ We,
                      const float* __restrict__ be, float* __restrict__ yeos) {
  int g = gtid(); if (g >= Gg) return;
  const float* xr = xpool + (size_t)g * HID;
  float s = 0.0f;
#pragma unroll
  for (int j = 0; j < HID; ++j) s += xr[j] * We[j];
  yeos[g] = s + be[0];
}

extern "C" void kernel_launch(void* const* d_in, const int* in_sizes, int n_in,
                              void* d_out, int out_size, void* d_ws, size_t ws_size,
                              hipStream_t stream) {
  const float* x      = (const float*)d_in[0];   // [N,128]
  const float* W_body = (const float*)d_in[1];   // [128,64]
  const float* b_body = (const float*)d_in[2];   // [64]
  const float* W_leaf = (const float*)d_in[3];   // [64,2]
  const float* b_leaf = (const float*)d_in[4];   // [2]
  const float* W_eos  = (const float*)d_in[5];   // [64,1]
  const float* b_eos  = (const float*)d_in[6];   // [1]
  const int*   src    = (const int*)d_in[7];     // [E]
  const int*   dst    = (const int*)d_in[8];     // [E]
  const int*   batch  = (const int*)d_in[9];     // [N]
  const int*   kpos   = (const int*)d_in[10];    // [N]

  float* out   = (float*)d_out;
  float* yleaf = out;                 // G*MAXN*2 = 262144
  float* yeos  = out + (size_t)Gg * MAXN * 2;

  // workspace layout (floats)
  float* ws    = (float*)d_ws;
  float* dinv  = ws;                                  // N       (deg -> dinv)
  float* hxw   = dinv + Nn;                           // N*64
  float* agg   = hxw + (size_t)Nn * HID;              // N*64    (-> h in place)
  float* tvec  = agg + (size_t)Nn * HID;              // N*2
  float* agg2  = tvec + (size_t)Nn * 2;               // N*2
  float* xpool = agg2 + (size_t)Nn * 2;               // G*64

  const int B = 256;
  auto cdiv = [](int a, int b) { return (a + b - 1) / b; };

  k_init<<<cdiv(Gg * MAXN * 2, B), B, 0, stream>>>(dinv, xpool, yleaf);
  k_deg<<<cdiv(Ee, B), B, 0, stream>>>(dst, dinv);
  k_dinv<<<cdiv(Nn, B), B, 0, stream>>>(dinv);

  // GEMM: N/16 = 6400 wave-strips (4 col tiles each), 8 waves per block
  k_gemm_wmma<<<cdiv(Nn / 16, 8), B, 0, stream>>>(x, W_body, dinv, b_body, hxw, agg);

  k_edge64<<<cdiv(Ee * 16, B), B, 0, stream>>>(hxw, dinv, src, dst, agg);
  k_leaky_pool<<<cdiv(Nn * HID, B), B, 0, stream>>>(agg, batch, xpool);

  k_leaf_xw<<<cdiv(Nn * 2, B), B, 0, stream>>>(agg, W_leaf, b_leaf, dinv, tvec, agg2);
  k_edge2<<<cdiv(Ee, B), B, 0, stream>>>(tvec, dinv, src, dst, agg2);
  k_scatter<<<cdiv(Nn, B), B, 0, stream>>>(agg2, batch, kpos, yleaf);
  k_eos<<<cdiv(Gg, B), B, 0, stream>>>(xpool, W_eos, b_eos, yeos);
}